// Schema_Relation_Network_6399501271451
// MI455X (gfx1250) — compile-verified
//
#include <hip/hip_runtime.h>
#include <hip/hip_bf16.h>

typedef __attribute__((ext_vector_type(2))) float v2f;
typedef __attribute__((ext_vector_type(8))) float v8f;

#define DIN 128
#define HF  64
#define RREL 3

// ---------- monotonic float<->uint key for atomicMax-based segment max ----------
__device__ __forceinline__ unsigned fkey(float f) {
  unsigned b = __float_as_uint(f);
  return (b & 0x80000000u) ? ~b : (b | 0x80000000u);
}
__device__ __forceinline__ float funkey(unsigned k) {
  unsigned b = (k & 0x80000000u) ? (k & 0x7FFFFFFFu) : ~k;
  return __uint_as_float(b);
}

// ---------- GEMM: H = X @ W + b via V_WMMA_F32_16X16X4_F32 ----------
// grid.y = 0 -> h_dst = dst_feat @ Wt[0] + bt[0]
// grid.y = r+1 -> h_src[r] = neigh[r] @ Wt[r+1] + bt[r+1]
// One wave computes a 16x16 tile; block = 4 waves = 4 column tiles (64 cols).
__global__ void gemm_h_kernel(const float* __restrict__ dst_feat,
                              const float* __restrict__ neigh,
                              const float* __restrict__ Wt,
                              const float* __restrict__ bt,
                              float* __restrict__ hdst,
                              float* __restrict__ hsrc,
                              int N, int Ns) {
  const int y = blockIdx.y;
  const float* X;
  float* Out;
  int rows;
  if (y == 0) { X = dst_feat; Out = hdst; rows = N; }
  else {
    X = neigh + (size_t)(y - 1) * Ns * DIN;
    Out = hsrc + (size_t)(y - 1) * Ns * HF;
    rows = Ns;
  }
  const float* W = Wt + (size_t)y * DIN * HF;

  const int lane = threadIdx.x & 31;
  const int wv   = threadIdx.x >> 5;   // 0..3 : column tile
  const int half = lane >> 4;          // 0/1 : K split within fragment
  const int lm   = lane & 15;

  const int row0 = blockIdx.x * 16;
  if (row0 >= rows) return;
  const int col = wv * 16 + lm;

  int arow = row0 + lm;
  if (arow >= rows) arow = rows - 1;   // clamp loads; stores guarded below
  const float* xrow = X + (size_t)arow * DIN + 2 * half;
  const float* wcol = W + (size_t)(2 * half) * HF + col;

  v8f c = {};
#pragma unroll
  for (int k = 0; k < DIN; k += 4) {
    v2f a = *(const v2f*)(xrow + k);       // A 16x4: lanes0-15 K=k..k+1, lanes16-31 K=k+2..k+3
    v2f b;
    b.x = wcol[(size_t)k * HF];            // B 4x16: same K split, N = lm per lane
    b.y = wcol[(size_t)(k + 1) * HF];
    c = __builtin_amdgcn_wmma_f32_16x16x4_f32(false, a, false, b, (short)0, c,
                                              false, false);
  }

  const float bias = bt[y * HF + col];
#pragma unroll
  for (int i = 0; i < 8; i++) {
    int m = row0 + i + 8 * half;           // D layout: VGPR i -> M=i (lanes<16), M=i+8 (lanes>=16)
    if (m < rows) Out[(size_t)m * HF + col] = c[i] + bias;
  }
}

// ---------- per-node attention logits: el[r][n], er[r][n] ----------
__global__ void elr_kernel(const float* __restrict__ hdst,
                           const float* __restrict__ hsrc,
                           const float* __restrict__ attn_l,
                           const float* __restrict__ attn_r,
                           float* __restrict__ el, float* __restrict__ er,
                           int N, int Ns) {
  const int r = blockIdx.y;
  const int n = blockIdx.x * 256 + threadIdx.x;
  const float* al = attn_l + r * HF;
  const float* ar = attn_r + r * HF;
  if (n < Ns) {
    const float* hs = hsrc + ((size_t)r * Ns + n) * HF;
    float acc = 0.f;
#pragma unroll 8
    for (int k = 0; k < HF; k++) acc += hs[k] * al[k];
    el[(size_t)r * Ns + n] = acc;
  }
  if (n < N) {
    const float* hd = hdst + (size_t)n * HF;
    float acc = 0.f;
#pragma unroll 8
    for (int k = 0; k < HF; k++) acc += hd[k] * ar[k];
    er[(size_t)r * N + n] = acc;
  }
}

// ---------- edge pass 1: e = leaky_relu(el[src]+er[dst]); segment max ----------
__global__ void edge_pass1_kernel(const int* __restrict__ src_idx,
                                  const int* __restrict__ dst_idx,
                                  const float* __restrict__ el,
                                  const float* __restrict__ er,
                                  float* __restrict__ ebuf,
                                  unsigned* __restrict__ mkey,
                                  int E, int N, int Ns) {
  const int r = blockIdx.y;
  const int i = blockIdx.x * 256 + threadIdx.x;
  if (i >= E) return;
  const int s = src_idx[(size_t)r * E + i];
  const int d = dst_idx[(size_t)r * E + i];
  float e = el[(size_t)r * Ns + s] + er[(size_t)r * N + d];
  e = (e > 0.f) ? e : 0.01f * e;
  ebuf[(size_t)r * E + i] = e;
  atomicMax(&mkey[(size_t)r * N + d], fkey(e));
}

// ---------- edge pass 2: a = exp(e - m[dst]); segment sum ----------
__global__ void edge_pass2_kernel(const int* __restrict__ dst_idx,
                                  const unsigned* __restrict__ mkey,
                                  float* __restrict__ ebuf,
                                  float* __restrict__ ssum,
                                  int E, int N) {
  const int r = blockIdx.y;
  const int i = blockIdx.x * 256 + threadIdx.x;
  if (i >= E) return;
  const int d = dst_idx[(size_t)r * E + i];
  const float m = funkey(mkey[(size_t)r * N + d]);
  const float a = __expf(ebuf[(size_t)r * E + i] - m);
  ebuf[(size_t)r * E + i] = a;
  unsafeAtomicAdd(&ssum[(size_t)r * N + d], a);
}

// ---------- edge pass 3: z_r[dst] += (a/s) * h_src[src]  (one wave per edge) ----------
__global__ void edge_pass3_kernel(const int* __restrict__ src_idx,
                                  const int* __restrict__ dst_idx,
                                  const float* __restrict__ ebuf,
                                  const float* __restrict__ ssum,
                                  const float* __restrict__ hsrc,
                                  float* __restrict__ zr,
                                  int E, int N, int Ns) {
  const int r = blockIdx.y;
  const int lane = threadIdx.x & 31;
  const int i = (int)((blockIdx.x * (unsigned)blockDim.x + threadIdx.x) >> 5);
  if (i >= E) return;
  const int s = src_idx[(size_t)r * E + i];
  const int d = dst_idx[(size_t)r * E + i];
  const float coef = ebuf[(size_t)r * E + i] / ssum[(size_t)r * N + d];
  const v2f hv = *(const v2f*)(hsrc + ((size_t)r * Ns + s) * HF + lane * 2);
  float* zp = zr + ((size_t)r * N + d) * HF + lane * 2;
  unsafeAtomicAdd(zp, hv.x * coef);
  unsafeAtomicAdd(zp + 1, hv.y * coef);
}

// ---------- elementwise ELU on z_r ----------
__global__ void elu_kernel(float* __restrict__ zr, size_t total) {
  size_t idx = (size_t)blockIdx.x * 256 + threadIdx.x;
  if (idx >= total) return;
  float x = zr[idx];
  zr[idx] = (x > 0.f) ? x : (__expf(x) - 1.f);
}

// ---------- semantic attention accumulation: wsum[r] += tanh(z @ W1 + b1) @ w2 ----------
__global__ void wsem_kernel(const float* __restrict__ zr,
                            const float* __restrict__ W1,
                            const float* __restrict__ b1,
                            const float* __restrict__ w2,
                            float* __restrict__ wsum, int N) {
  const int r = blockIdx.y;
  const int n = blockIdx.x * 128 + threadIdx.x;
  float acc = 0.f;
  if (n < N) {
    const float* z = zr + ((size_t)r * N + n) * HF;
    float zreg[HF];
#pragma unroll
    for (int k = 0; k < HF; k++) zreg[k] = z[k];
    for (int j = 0; j < 128; j++) {
      float t = b1[j];
#pragma unroll
      for (int k = 0; k < HF; k++) t += zreg[k] * W1[k * 128 + j];
      acc += tanhf(t) * w2[j];
    }
  }
  // wave32 shuffle reduce, then one atomic per wave
#pragma unroll
  for (int off = 16; off > 0; off >>= 1) acc += __shfl_down(acc, off, 32);
  if ((threadIdx.x & 31) == 0) unsafeAtomicAdd(&wsum[r], acc);
}

// ---------- mean + 3-way softmax over relations ----------
__global__ void softmax_kernel(const float* __restrict__ wsum,
                               float* __restrict__ aw, int N) {
  if (threadIdx.x == 0) {
    float inv = 1.f / (float)N;
    float w0 = wsum[0] * inv, w1 = wsum[1] * inv, w2v = wsum[2] * inv;
    float m = fmaxf(w0, fmaxf(w1, w2v));
    float e0 = __expf(w0 - m), e1 = __expf(w1 - m), e2 = __expf(w2v - m);
    float s = e0 + e1 + e2;
    aw[0] = e0 / s; aw[1] = e1 / s; aw[2] = e2 / s;
  }
}

// ---------- final mix: out = sum_r aw[r] * z_r ----------
__global__ void mix_kernel(const float* __restrict__ zr,
                           const float* __restrict__ aw,
                           float* __restrict__ out, int N) {
  size_t idx = (size_t)blockIdx.x * 256 + threadIdx.x;
  size_t total = (size_t)N * HF;
  if (idx >= total) return;
  out[idx] = aw[0] * zr[idx] + aw[1] * zr[total + idx] + aw[2] * zr[2 * total + idx];
}

extern "C" void kernel_launch(void* const* d_in, const int* in_sizes, int n_in,
                              void* d_out, int out_size, void* d_ws, size_t ws_size,
                              hipStream_t stream) {
  const float* dst_feat = (const float*)d_in[0];
  const float* neigh    = (const float*)d_in[1];
  const float* Wt       = (const float*)d_in[2];
  const float* bt       = (const float*)d_in[3];
  const float* attn_l   = (const float*)d_in[4];
  const float* attn_r   = (const float*)d_in[5];
  const float* W1       = (const float*)d_in[6];
  const float* b1       = (const float*)d_in[7];
  const float* w2       = (const float*)d_in[8];
  const int*   src_idx  = (const int*)d_in[9];
  const int*   dst_idx  = (const int*)d_in[10];
  float* out = (float*)d_out;

  const int N  = in_sizes[0] / DIN;
  const int Ns = in_sizes[1] / (RREL * DIN);
  const int E  = in_sizes[9] / RREL;

  // ---- workspace layout (floats) ----
  float* ws = (float*)d_ws;
  float*    hdst = ws;                                  // N*64
  float*    hsrc = hdst + (size_t)N * HF;               // 3*Ns*64
  float*    zr   = hsrc + (size_t)RREL * Ns * HF;       // 3*N*64
  float*    el   = zr + (size_t)RREL * N * HF;          // 3*Ns
  float*    er   = el + (size_t)RREL * Ns;              // 3*N
  float*    ssum = er + (size_t)RREL * N;               // 3*N
  unsigned* mkey = (unsigned*)(ssum + (size_t)RREL * N);// 3*N (0x0 == -inf key)
  float*    ebuf = (float*)(mkey + (size_t)RREL * N);   // 3*E
  float*    wsum = ebuf + (size_t)RREL * E;             // 3
  float*    aw   = wsum + RREL;                         // 3

  // ---- zero-init accumulators (graph-capturable memsets) ----
  hipMemsetAsync(zr, 0, (size_t)RREL * N * HF * sizeof(float), stream);
  hipMemsetAsync(ssum, 0, (size_t)2 * RREL * N * sizeof(float), stream); // ssum + mkey
  hipMemsetAsync(wsum, 0, 2 * RREL * sizeof(float), stream);             // wsum + aw

  // ---- 1) projections via f32 WMMA ----
  {
    int rowsMax = (N > Ns) ? N : Ns;
    dim3 grid((rowsMax + 15) / 16, RREL + 1);
    gemm_h_kernel<<<grid, 128, 0, stream>>>(dst_feat, neigh, Wt, bt, hdst, hsrc, N, Ns);
  }
  // ---- 2) per-node logits ----
  {
    int nMax = (N > Ns) ? N : Ns;
    dim3 grid((nMax + 255) / 256, RREL);
    elr_kernel<<<grid, 256, 0, stream>>>(hdst, hsrc, attn_l, attn_r, el, er, N, Ns);
  }
  // ---- 3) edge softmax ----
  {
    dim3 grid((E + 255) / 256, RREL);
    edge_pass1_kernel<<<grid, 256, 0, stream>>>(src_idx, dst_idx, el, er, ebuf, mkey, E, N, Ns);
    edge_pass2_kernel<<<grid, 256, 0, stream>>>(dst_idx, mkey, ebuf, ssum, E, N);
  }
  // ---- 4) weighted aggregation: one wave per edge ----
  {
    dim3 grid((E + 7) / 8, RREL);
    edge_pass3_kernel<<<grid, 256, 0, stream>>>(src_idx, dst_idx, ebuf, ssum, hsrc, zr, E, N, Ns);
  }
  // ---- 5) ELU ----
  {
    size_t total = (size_t)RREL * N * HF;
    elu_kernel<<<(unsigned)((total + 255) / 256), 256, 0, stream>>>(zr, total);
  }
  // ---- 6) semantic attention weights ----
  {
    dim3 grid((N + 127) / 128, RREL);
    wsem_kernel<<<grid, 128, 0, stream>>>(zr, W1, b1, w2, wsum, N);
    softmax_kernel<<<1, 32, 0, stream>>>(wsum, aw, N);
  }
  // ---- 7) final mix ----
  {
    size_t total = (size_t)N * HF;
    mix_kernel<<<(unsigned)((total + 255) / 256), 256, 0, stream>>>(zr, aw, out, N);
  }
}